// DropBlock_5669356833657
// MI455X (gfx1250) — compile-verified
//
#include <hip/hip_runtime.h>
#include <stdint.h>

#define BS 7
#define HM 50
#define WM 50
#define HH 56
#define WW 56
#define PLANES (64 * 256)
#define NROWS (64 * 256 * 56)          // 917504 output rows
#define COUNT_M 51380224.0f            // 64*256*56*56

// ---------------------------------------------------------------------------
// Kernel A: one block per (b,c) plane.
//  - async global->LDS stage of the 50x50 u tile (CDNA5 async path)
//  - wave32 ballots -> 50 seed bit-rows (u64)
//  - horizontal 7-OR via shifts, vertical 7-OR via row words
//  - stash each row's keep-mask u64 at the start of that row in d_out
//  - popcount -> global atomic counter (in d_ws)
// ---------------------------------------------------------------------------
__global__ __launch_bounds__(256) void dropblock_mask_kernel(
    const float* __restrict__ u, const float* __restrict__ gamma,
    float* __restrict__ out, int* __restrict__ gcount)
{
    __shared__ __align__(16) float su[HM * WM];   // 10000 B
    __shared__ unsigned long long srow[HM];       // seed rows, then hdil in place
    __shared__ int s_cnt;

    const int t    = threadIdx.x;
    const int lane = t & 31;
    const int wid  = t >> 5;
    const int plane = blockIdx.x;

    if (t == 0) s_cnt = 0;

    // ---- Stage u plane into LDS with async global->LDS B128 copies ----
    {
        const unsigned long long gbase =
            (unsigned long long)(uintptr_t)u + (unsigned long long)plane * (HM * WM * 4);
        const unsigned lbase = (unsigned)(uintptr_t)(&su[0]); // flat->LDS: low 32 bits
        // 10000 B = 625 x 16 B chunks
        for (int c = t; c < (HM * WM * 4) / 16; c += 256) {
            unsigned long long ga = gbase + (unsigned long long)c * 16u;
            unsigned           la = lbase + (unsigned)c * 16u;
            asm volatile("global_load_async_to_lds_b128 %0, %1, off"
                         :: "v"(la), "v"(ga) : "memory");
        }
        asm volatile("s_wait_asynccnt 0" ::: "memory");
    }
    __syncthreads();

    const float g = gamma[0];

    // ---- Seed bit-rows via wave32 ballots (row r -> one u64, bits 0..49) ----
    for (int r = wid; r < HM; r += 8) {
        bool p0 = su[r * WM + lane] < g;
        unsigned m0 = __builtin_amdgcn_ballot_w32(p0);
        int c2 = lane + 32;
        bool p1 = (c2 < WM) && (su[r * WM + c2] < g);
        unsigned m1 = __builtin_amdgcn_ballot_w32(p1);
        if (lane == 0)
            srow[r] = (unsigned long long)m0 | ((unsigned long long)m1 << 32);
    }
    __syncthreads();

    // ---- Horizontal dilation: bit w of result = OR of seed bits w-6..w ----
    if (t < HM) {
        unsigned long long s = srow[t];
        srow[t] = s | (s << 1) | (s << 2) | (s << 3) | (s << 4) | (s << 5) | (s << 6);
    }
    __syncthreads();

    // ---- Vertical dilation + invert; stash keep-mask word per output row ----
    if (t < HH) {
        int lo = t - (BS - 1); if (lo < 0) lo = 0;
        int hi = t;            if (hi > HM - 1) hi = HM - 1;
        unsigned long long d = 0ull;
        for (int r = lo; r <= hi; ++r) d |= srow[r];
        unsigned long long keep = (~d) & ((1ull << WW) - 1ull);

        char* dst = (char*)out + ((size_t)plane * (HH * WW) + (size_t)t * WW) * 4;
        __builtin_memcpy(dst, &keep, 8);   // 8-byte aligned (row stride 224 B)

        atomicAdd(&s_cnt, __popcll(keep));
    }
    __syncthreads();
    if (t == 0) atomicAdd(gcount, s_cnt);
}

// ---------------------------------------------------------------------------
// Kernel B: one wave per output row (56 floats).
// Load the stashed u64 row-mask first (register dependency orders it before
// the stores that overwrite it), then out = x * bit * (countM / count_ones).
// ---------------------------------------------------------------------------
__global__ __launch_bounds__(256) void dropblock_apply_kernel(
    const float* __restrict__ x, float* out, const int* __restrict__ gcount)
{
    const int lane = threadIdx.x & 31;
    const int wid  = threadIdx.x >> 5;
    const size_t row  = (size_t)blockIdx.x * 8 + wid;  // grid sized exactly
    const size_t base = row * WW;

    unsigned long long m;
    __builtin_memcpy(&m, (const char*)out + base * 4, 8);

    const float sc = COUNT_M / (float)(*gcount);

    {
        float xv = x[base + lane];
        out[base + lane] = ((m >> lane) & 1ull) ? xv * sc : 0.0f;
    }
    int c2 = lane + 32;
    if (c2 < WW) {
        float xv = x[base + c2];
        out[base + c2] = ((m >> c2) & 1ull) ? xv * sc : 0.0f;
    }
}

extern "C" void kernel_launch(void* const* d_in, const int* in_sizes, int n_in,
                              void* d_out, int out_size, void* d_ws, size_t ws_size,
                              hipStream_t stream)
{
    (void)in_sizes; (void)n_in; (void)out_size; (void)ws_size;
    const float* x     = (const float*)d_in[0];
    const float* u     = (const float*)d_in[1];
    const float* gamma = (const float*)d_in[2];
    float* out  = (float*)d_out;
    int* gcount = (int*)d_ws;              // 4-byte global ones-counter

    hipMemsetAsync(gcount, 0, sizeof(int), stream);
    dropblock_mask_kernel<<<PLANES, 256, 0, stream>>>(u, gamma, out, gcount);
    dropblock_apply_kernel<<<NROWS / 8, 256, 0, stream>>>(x, out, gcount);
}